// BuildingGCN_49185965474186
// MI455X (gfx1250) — compile-verified
//
#include <hip/hip_runtime.h>
#include <hip/hip_bf16.h>

#define N_NODES  100000
#define N_EDGES  3200000
#define N_GRAPHS 256
#define HID      64

typedef float v2f __attribute__((ext_vector_type(2)));
typedef float v8f __attribute__((ext_vector_type(8)));

// ---------------------------------------------------------------------------
// WMMA fp32 GEMM + fused self-loop epilogue:
//   Y[N,M]     = X[N,K] @ W[K,M]
//   Binit[N,M] = Y * recip[row]        (GCN self-loop term, seeds scatter-add)
// One wave32 per 16-row x M-col strip; A fragment reused across M/16 column
// tiles.  4 waves / 128-thread block, wave-uniform tile pick (EXEC all-ones
// per wave, as WMMA requires).  V_WMMA_F32_16X16X4_F32 layouts per ISA 7.12.2.
// ---------------------------------------------------------------------------
template <int K, int M, bool FUSE_SEED>
__global__ __launch_bounds__(128)
void gemm_wmma_f32(const float* __restrict__ X, const float* __restrict__ W,
                   float* __restrict__ Y, float* __restrict__ Binit,
                   const float* __restrict__ recip, int nRowTiles) {
  constexpr int NT = M / 16;                 // column tiles per wave
  const int lane = threadIdx.x & 31;
  const int wave = threadIdx.x >> 5;
  const int tile = blockIdx.x * 4 + wave;    // wave-uniform
  if (tile >= nRowTiles) return;

  const int half = lane >> 4;                // K-pair select
  const int l    = lane & 15;
  const int row  = tile * 16 + l;

  v8f acc[NT];
#pragma unroll
  for (int t = 0; t < NT; ++t) acc[t] = (v8f){};

  for (int k0 = 0; k0 < K; k0 += 4) {
    const int ka = k0 + half * 2;
    v2f a;
    a.x = X[(size_t)row * K + ka];
    a.y = X[(size_t)row * K + ka + 1];
#pragma unroll
    for (int t = 0; t < NT; ++t) {
      const int col = t * 16 + l;
      v2f b;
      b.x = W[(size_t)ka * M + col];
      b.y = W[(size_t)(ka + 1) * M + col];
      acc[t] = __builtin_amdgcn_wmma_f32_16x16x4_f32(
          false, a, false, b, (short)0, acc[t], false, false);
    }
  }

  float rc[8];
  if (FUSE_SEED) {
#pragma unroll
    for (int r = 0; r < 8; ++r) rc[r] = recip[tile * 16 + half * 8 + r];
  }
#pragma unroll
  for (int t = 0; t < NT; ++t) {
    const int col = t * 16 + l;
#pragma unroll
    for (int r = 0; r < 8; ++r) {
      const int orow = tile * 16 + half * 8 + r;
      const float v = acc[t][r];
      Y[(size_t)orow * M + col] = v;
      if (FUSE_SEED) Binit[(size_t)orow * M + col] = v * rc[r];
    }
  }
}

// ---------------------------------------------------------------------------
// Degree / normalization
// ---------------------------------------------------------------------------
__global__ void fill_f32(float* __restrict__ p, float v, long long n) {
  long long i = (long long)blockIdx.x * blockDim.x + threadIdx.x;
  long long s = (long long)gridDim.x * blockDim.x;
  for (; i < n; i += s) p[i] = v;
}

__global__ void edge_deg(const int* __restrict__ dst, float* __restrict__ deg,
                         int nE) {
  int i = blockIdx.x * blockDim.x + threadIdx.x;
  int s = gridDim.x * blockDim.x;
  for (; i < nE; i += s) atomicAdd(&deg[dst[i]], 1.0f);
}

__global__ void finish_deg(const float* __restrict__ deg,
                           float* __restrict__ dinv,
                           float* __restrict__ recip, int n) {
  int i = blockIdx.x * blockDim.x + threadIdx.x;
  int s = gridDim.x * blockDim.x;
  for (; i < n; i += s) {
    float d = deg[i];
    dinv[i]  = __frsqrt_rn(d);
    recip[i] = 1.0f / d;
  }
}

// Fallback seed pass (only when workspace can't hold 3 feature buffers):
// B[i,c] = A[i,c] * recip[i]
template <int M>
__global__ void seed_self(const float* __restrict__ A, float* __restrict__ B,
                          const float* __restrict__ recip) {
  constexpr int SH = (M == 128) ? 7 : 6;
  long long n = (long long)N_NODES * M;
  long long t = (long long)blockIdx.x * blockDim.x + threadIdx.x;
  long long s = (long long)gridDim.x * blockDim.x;
  for (; t < n; t += s) B[t] = A[t] * recip[(int)(t >> SH)];
}

// ---------------------------------------------------------------------------
// Scatter-add:  B[dst] += dinv[src]*dinv[dst] * A[src]
// float4 per thread; consecutive threads cover consecutive channel chunks of
// one edge -> coalesced gathers.  M is a compile-time power of two so the
// edge / chunk split is a shift+mask (this loop runs ~100M times for M=128).
// A/B (<=51MB each) are L2-resident, so atomics resolve in the 192MB L2.
// ---------------------------------------------------------------------------
template <int M>
__global__ void edge_agg(const float* __restrict__ A, float* __restrict__ B,
                         const int* __restrict__ src, const int* __restrict__ dst,
                         const float* __restrict__ dinv) {
  constexpr int CH  = M >> 2;                 // float4 chunks per edge
  constexpr int LSH = (M == 128) ? 5 : 4;     // log2(CH)
  long long total = (long long)N_EDGES * CH;
  long long t = (long long)blockIdx.x * blockDim.x + threadIdx.x;
  long long s = (long long)gridDim.x * blockDim.x;
  for (; t < total; t += s) {
    int e  = (int)(t >> LSH);
    int c4 = ((int)t & (CH - 1)) << 2;
    int sn = src[e];
    int dn = dst[e];
    float norm = dinv[sn] * dinv[dn];
    const float4 v = *(const float4*)(A + (size_t)sn * M + c4);
    float* out = B + (size_t)dn * M + c4;
    atomicAdd(out + 0, norm * v.x);
    atomicAdd(out + 1, norm * v.y);
    atomicAdd(out + 2, norm * v.z);
    atomicAdd(out + 3, norm * v.w);
  }
}

template <int M>
__global__ void bias_relu(float* __restrict__ B, const float* __restrict__ bias) {
  long long n = (long long)N_NODES * M;
  long long t = (long long)blockIdx.x * blockDim.x + threadIdx.x;
  long long s = (long long)gridDim.x * blockDim.x;
  for (; t < n; t += s) {
    int c = (int)t & (M - 1);
    B[t] = fmaxf(B[t] + bias[c], 0.0f);
  }
}

// ---------------------------------------------------------------------------
// Global mean pool (fused layer-3 bias+ReLU), then MLP head
// ---------------------------------------------------------------------------
__global__ void pool_bias_relu(const float* __restrict__ h,
                               const float* __restrict__ bias,
                               const int* __restrict__ batch,
                               float* __restrict__ gsum, float* __restrict__ gcnt) {
  long long n = (long long)N_NODES * HID;
  long long t = (long long)blockIdx.x * blockDim.x + threadIdx.x;
  long long s = (long long)gridDim.x * blockDim.x;
  for (; t < n; t += s) {
    int i = (int)(t >> 6);      // HID == 64
    int c = (int)t & 63;
    int b = batch[i];
    float v = fmaxf(h[t] + bias[c], 0.0f);
    atomicAdd(&gsum[(size_t)b * HID + c], v);
    if (c == 0) atomicAdd(&gcnt[b], 1.0f);
  }
}

__global__ void head_mlp(const float* __restrict__ gsum, const float* __restrict__ gcnt,
                         const float* __restrict__ Wl1, const float* __restrict__ bl1,
                         const float* __restrict__ Wl2, const float* __restrict__ bl2,
                         float* __restrict__ out) {
  int g = blockIdx.x * blockDim.x + threadIdx.x;
  if (g >= N_GRAPHS) return;
  float inv = 1.0f / fmaxf(gcnt[g], 1.0f);
  float gv[HID];
#pragma unroll
  for (int c = 0; c < HID; ++c) gv[c] = gsum[(size_t)g * HID + c] * inv;
  float o = bl2[0];
  for (int j = 0; j < HID / 2; ++j) {
    float hv = bl1[j];
#pragma unroll
    for (int c = 0; c < HID; ++c) hv += gv[c] * Wl1[c * (HID / 2) + j];
    o += fmaxf(hv, 0.0f) * Wl2[j];
  }
  out[g] = o;
}

// ---------------------------------------------------------------------------
extern "C" void kernel_launch(void* const* d_in, const int* in_sizes, int n_in,
                              void* d_out, int out_size, void* d_ws, size_t ws_size,
                              hipStream_t stream) {
  const float* x    = (const float*)d_in[0];
  const int*   eidx = (const int*)d_in[1];
  const int*   bat  = (const int*)d_in[2];
  const float* W1 = (const float*)d_in[3];  const float* b1 = (const float*)d_in[4];
  const float* W2 = (const float*)d_in[5];  const float* b2 = (const float*)d_in[6];
  const float* W3 = (const float*)d_in[7];  const float* b3 = (const float*)d_in[8];
  const float* Wl1 = (const float*)d_in[9];  const float* bl1 = (const float*)d_in[10];
  const float* Wl2 = (const float*)d_in[11]; const float* bl2 = (const float*)d_in[12];
  float* out = (float*)d_out;

  const int* src = eidx;
  const int* dst = eidx + N_EDGES;

  // Workspace: prefer 3 feature buffers [N,128] (fully fused epilogues,
  // ~156 MB); fall back to 2 buffers + seed pass (~105 MB).  ws_size is a
  // fixed host value, so this choice is deterministic across calls.
  const size_t FB    = (size_t)N_NODES * 128;           // floats per buffer
  const size_t tailF = 3 * (size_t)N_NODES + (size_t)N_GRAPHS * HID + N_GRAPHS;
  const bool triple  = ws_size >= (3 * FB + tailF) * sizeof(float);

  float* A = (float*)d_ws;
  float* B = A + FB;
  float* C = triple ? (B + FB) : B;          // C aliases B in fallback mode
  float* deg   = (triple ? C + FB : B + FB);
  float* dinv  = deg + N_NODES;
  float* recip = dinv + N_NODES;
  float* gsum  = recip + N_NODES;
  float* gcnt  = gsum + (size_t)N_GRAPHS * HID;

  const int TB = 256;
  const int nodeBlocks = (N_NODES + TB - 1) / TB;
  const int edgeBlocks = (N_EDGES + TB - 1) / TB;

  // --- degree / normalization ---
  fill_f32<<<nodeBlocks, TB, 0, stream>>>(deg, 1.0f, N_NODES);   // self-loop
  edge_deg<<<edgeBlocks, TB, 0, stream>>>(dst, deg, N_EDGES);
  finish_deg<<<nodeBlocks, TB, 0, stream>>>(deg, dinv, recip, N_NODES);

  const int rowTiles   = N_NODES / 16;              // 6250 exactly
  const int gemmBlocks = (rowTiles + 3) / 4;        // 4 row-tile waves / block

  const long long nm64  = (long long)N_NODES * 64;
  const long long nm128 = (long long)N_NODES * 128;
  const int nm64B  = (int)((nm64  + TB - 1) / TB);
  const int nm128B = (int)((nm128 + TB - 1) / TB);
  const int ec64B  = (int)(((long long)N_EDGES * 16 + TB - 1) / TB);
  const int ec128B = (int)(((long long)N_EDGES * 32 + TB - 1) / TB);

  // --- layer 1: x[N,8] @ W1[8,64] -> h1 in B ---
  gemm_wmma_f32<8, 64, true><<<gemmBlocks, 128, 0, stream>>>(x, W1, A, B,
                                                             recip, rowTiles);
  edge_agg<64><<<ec64B, TB, 0, stream>>>(A, B, src, dst, dinv);
  bias_relu<64><<<nm64B, TB, 0, stream>>>(B, b1);

  // --- layer 2: h1[N,64] @ W2[64,128] -> h2 in C ---
  if (triple) {
    gemm_wmma_f32<64, 128, true><<<gemmBlocks, 128, 0, stream>>>(B, W2, A, C,
                                                                 recip, rowTiles);
  } else {
    gemm_wmma_f32<64, 128, false><<<gemmBlocks, 128, 0, stream>>>(B, W2, A, A,
                                                                  recip, rowTiles);
    seed_self<128><<<nm128B, TB, 0, stream>>>(A, C, recip);
  }
  edge_agg<128><<<ec128B, TB, 0, stream>>>(A, C, src, dst, dinv);
  bias_relu<128><<<nm128B, TB, 0, stream>>>(C, b2);

  // --- layer 3: h2[N,128] @ W3[128,64] -> h3 in B ---
  if (triple) {
    gemm_wmma_f32<128, 64, true><<<gemmBlocks, 128, 0, stream>>>(C, W3, A, B,
                                                                 recip, rowTiles);
  } else {
    gemm_wmma_f32<128, 64, false><<<gemmBlocks, 128, 0, stream>>>(C, W3, A, A,
                                                                  recip, rowTiles);
    seed_self<64><<<nm64B, TB, 0, stream>>>(A, B, recip);
  }
  edge_agg<64><<<ec64B, TB, 0, stream>>>(A, B, src, dst, dinv);

  // --- global mean pool (fused layer-3 bias+ReLU) + head ---
  fill_f32<<<(N_GRAPHS * HID + TB - 1) / TB, TB, 0, stream>>>(gsum, 0.0f,
                                                              N_GRAPHS * HID);
  fill_f32<<<1, TB, 0, stream>>>(gcnt, 0.0f, N_GRAPHS);
  pool_bias_relu<<<nm64B, TB, 0, stream>>>(B, b3, bat, gsum, gcnt);
  head_mlp<<<(N_GRAPHS + TB - 1) / TB, TB, 0, stream>>>(gsum, gcnt, Wl1, bl1,
                                                        Wl2, bl2, out);
}